// EventVoxelizer_60172491817011
// MI455X (gfx1250) — compile-verified
//
#include <hip/hip_runtime.h>
#include <stdint.h>

// Problem geometry (fixed by the reference)
#define THRESH 3.0f
#define MAXPTS 200000
#define NB 4
#define NS 8
#define NC 8            // 2*4 event channels
#define NH 480
#define NW 640
#define HW (NH * NW)            // 307200
#define NFR (NB * NS)           // 32
#define NPIX (NFR * HW)         // 9830400
#define TPB 256
#define PPB 1024                // pixels per block (divides HW: 307200/1024 = 300)
#define NBLK (NPIX / PPB)       // 9600
#define WPB (PPB / 32)          // 32 ballot words per block

#if defined(__gfx1250__) && __has_builtin(__builtin_amdgcn_global_load_async_to_lds_b128)
#define HAVE_ASYNC_LDS 1
#else
#define HAVE_ASYNC_LDS 0
#endif

typedef __attribute__((ext_vector_type(4))) int v4i_t;
typedef __attribute__((address_space(1))) v4i_t* gv4i_p;   // global
typedef __attribute__((address_space(3))) v4i_t* lv4i_p;   // LDS

static __device__ __forceinline__ unsigned ballot32(bool p) {
#if __has_builtin(__builtin_amdgcn_ballot_w32)
  return __builtin_amdgcn_ballot_w32(p);
#else
  return (unsigned)__ballot(p);
#endif
}

static __device__ __forceinline__ void wait_async0() {
#if __has_builtin(__builtin_amdgcn_s_wait_asynccnt)
  __builtin_amdgcn_s_wait_asynccnt(0);
#else
  asm volatile("s_wait_asynccnt 0" ::: "memory");
#endif
}

// ---------------- Pass 1: mask + per-block counts -------------------------
__global__ __launch_bounds__(TPB) void ev_pass1_count(
    const float* __restrict__ frames,
    unsigned* __restrict__ ballotWords,   // NPIX/32 words
    unsigned* __restrict__ blkCounts) {   // NBLK
  __shared__ float tile[NC * PPB];        // 32 KB
  __shared__ unsigned s_cnt;
  const int t = threadIdx.x;
  const int blk = blockIdx.x;
  if (t == 0) s_cnt = 0;

  const int pix0 = blk * PPB;             // block fully inside one bf frame
  const int bf = pix0 / HW;
  const int p0 = pix0 - bf * HW;
  const float* fbase = frames + (size_t)bf * NC * HW + p0;

  // Stage 8 channel rows (4KB each) into LDS.
#if HAVE_ASYNC_LDS
#pragma unroll
  for (int c = 0; c < NC; ++c) {
    float* g = const_cast<float*>(fbase + (size_t)c * HW + t * 4);
    __builtin_amdgcn_global_load_async_to_lds_b128(
        (gv4i_p)g, (lv4i_p)&tile[c * PPB + t * 4], 0, 0);
  }
  wait_async0();
#else
#pragma unroll
  for (int c = 0; c < NC; ++c) {
    const float4 v = *(const float4*)(fbase + (size_t)c * HW + t * 4);
    *(float4*)(&tile[c * PPB + t * 4]) = v;
  }
#endif
  __syncthreads();

  const int lane = t & 31;
#pragma unroll
  for (int k = 0; k < PPB / TPB; ++k) {
    const int j = k * TPB + t;            // pixel index within block
    float m = 0.0f;
#pragma unroll
    for (int c = 0; c < NC; ++c) m = fmaxf(m, fabsf(tile[c * PPB + j]));
    const bool pred = m > THRESH;
    const unsigned word = ballot32(pred);
    if (lane == 0) {
      ballotWords[(size_t)blk * WPB + k * (TPB / 32) + (t >> 5)] = word;
      atomicAdd(&s_cnt, (unsigned)__popc(word));
    }
  }
  __syncthreads();
  if (t == 0) blkCounts[blk] = s_cnt;
}

// ---------------- Pass 2: exclusive scan of block counts ------------------
__global__ __launch_bounds__(256) void ev_pass2_scan(
    const unsigned* __restrict__ counts, unsigned* __restrict__ offsets) {
  __shared__ unsigned sdata[256];
  __shared__ unsigned srun;
  const int t = threadIdx.x;
  if (t == 0) srun = 0;
  __syncthreads();
  for (int base = 0; base < NBLK; base += 256) {
    const int i = base + t;
    const unsigned v = (i < NBLK) ? counts[i] : 0u;
    sdata[t] = v;
    __syncthreads();
    for (int off = 1; off < 256; off <<= 1) {
      const unsigned add = (t >= off) ? sdata[t - off] : 0u;
      __syncthreads();
      sdata[t] += add;
      __syncthreads();
    }
    const unsigned incl = sdata[t];
    const unsigned run = srun;
    if (i < NBLK) offsets[i] = run + incl - v;   // exclusive prefix
    __syncthreads();
    if (t == 255) srun = run + sdata[255];
    __syncthreads();
  }
}

// ---------------- Pass 3: gather + ordered scatter -------------------------
__global__ __launch_bounds__(TPB) void ev_pass3_emit(
    const float* __restrict__ frames,
    const unsigned* __restrict__ ballotWords,
    const unsigned* __restrict__ blkOffsets,
    float* __restrict__ featOut,          // MAXPTS x 8
    int* __restrict__ coordOut) {         // MAXPTS x 4
  __shared__ unsigned words[WPB];
  __shared__ unsigned wpref[WPB];
  const int t = threadIdx.x;
  const int blk = blockIdx.x;
  if (t < WPB) words[t] = ballotWords[(size_t)blk * WPB + t];
  __syncthreads();
  if (t == 0) {
    unsigned run = 0;
    for (int i = 0; i < WPB; ++i) { wpref[i] = run; run += (unsigned)__popc(words[i]); }
  }
  __syncthreads();

  const unsigned base = blkOffsets[blk];
  if (base >= MAXPTS) return;             // whole block beyond the cap

  const int pix0 = blk * PPB;
  const int bf = pix0 / HW;
  const int p0 = pix0 - bf * HW;
  const int lane = t & 31;
  const int w = t >> 5;
  const int batch = bf >> 3;              // bf / S
  const int timei = bf & 7;               // bf % S

#pragma unroll
  for (int k = 0; k < PPB / TPB; ++k) {
    const int widx = k * (TPB / 32) + w;
    const unsigned mask = words[widx];
    if ((mask >> lane) & 1u) {
      const unsigned rank =
          base + wpref[widx] + (unsigned)__popc(mask & ((1u << lane) - 1u));
      if (rank < MAXPTS) {
        const int p = p0 + k * TPB + t;
        const int y = p / NW;
        const int x = p - y * NW;
        const float* src = frames + (size_t)bf * NC * HW + p;
        float* fo = featOut + (size_t)rank * NC;
#pragma unroll
        for (int c = 0; c < NC; ++c) fo[c] = src[(size_t)c * HW];
        int* co = coordOut + (size_t)rank * 4;
        co[0] = batch; co[1] = timei; co[2] = y; co[3] = x;
      }
    }
  }
}

// ---------------- zero fill for the padded tail ----------------------------
__global__ void ev_zero(unsigned* __restrict__ out, int n) {
  const int i = blockIdx.x * blockDim.x + threadIdx.x;
  if (i < n) out[i] = 0u;
}

extern "C" void kernel_launch(void* const* d_in, const int* in_sizes, int n_in,
                              void* d_out, int out_size, void* d_ws, size_t ws_size,
                              hipStream_t stream) {
  (void)in_sizes; (void)n_in; (void)ws_size;
  const float* frames = (const float*)d_in[0];

  // d_out = features (MAXPTS*8 f32) then coords (MAXPTS*4 i32), concatenated.
  float* featOut = (float*)d_out;
  int* coordOut = (int*)((float*)d_out + (size_t)MAXPTS * NC);

  // workspace carve-up (~1.3 MB)
  unsigned* ballotWords = (unsigned*)d_ws;            // NPIX/32 = 307200 words
  unsigned* blkCounts = ballotWords + (NPIX / 32);    // NBLK
  unsigned* blkOffsets = blkCounts + NBLK;            // NBLK

  ev_zero<<<(out_size + 255) / 256, 256, 0, stream>>>((unsigned*)d_out, out_size);
  ev_pass1_count<<<NBLK, TPB, 0, stream>>>(frames, ballotWords, blkCounts);
  ev_pass2_scan<<<1, 256, 0, stream>>>(blkCounts, blkOffsets);
  ev_pass3_emit<<<NBLK, TPB, 0, stream>>>(frames, ballotWords, blkOffsets,
                                          featOut, coordOut);
}